// VectorQuantizer_41171556499967
// MI455X (gfx1250) — compile-verified
//
#include <hip/hip_runtime.h>
#include <hip/hip_bf16.h>
#include <stdint.h>

#define K_CODES 512
#define DIM 64
#define N_VECS 65536
#define ROWS_PER_WG 128
#define LDS_STRIDE 68  // 64 + pad: 16B-aligned rows (272B), conflict-free strided reads

typedef __attribute__((ext_vector_type(2))) float v2f;
typedef __attribute__((ext_vector_type(8))) float v8f;

// ---------------------------------------------------------------------------
// zero the scratch: ws[0] = sse accumulator (float), ws[1..512] = counts (u32)
// ---------------------------------------------------------------------------
__global__ void vq_init(float* ws) {
  int i = blockIdx.x * blockDim.x + threadIdx.x;
  if (i < 1 + K_CODES) ws[i] = 0.0f;  // bit pattern 0 is also uint 0
}

// ---------------------------------------------------------------------------
// main kernel: one WG = 128 rows of flat(z_e), all 512 codes
// ---------------------------------------------------------------------------
__global__ __launch_bounds__(256) void vq_main(
    const float* __restrict__ z_e, const float* __restrict__ emb,
    float* __restrict__ out, float* __restrict__ out_idx,
    float* __restrict__ sse_g, unsigned int* __restrict__ counts) {
  __shared__ float eL[K_CODES * LDS_STRIDE];
  __shared__ float e2s[K_CODES];
  __shared__ int sIdx[ROWS_PER_WG];
  __shared__ float wsum[8];

  const int tid = threadIdx.x;
  const int lane = tid & 31;
  const int wave = tid >> 5;
  const int l15 = lane & 15;
  const int lh = lane >> 4;

  const int wg = blockIdx.x;
  const int b = wg >> 5;  // 32 WGs per batch image (4096 / 128)
  const int hw_base = (wg * ROWS_PER_WG) & 4095;
  const size_t zbatch = (size_t)b * (DIM * 4096);

  // ---- stage codebook into LDS (padded rows) via async global->LDS DMA ----
  // Each thread copies 16B chunks. generic-ptr low 32 bits == LDS byte offset.
  for (int G = tid; G < K_CODES * (DIM / 4); G += 256) {
    int n = G >> 4;
    int koff = (G & 15) << 2;
    unsigned lds_off = (unsigned)(uintptr_t)&eL[n * LDS_STRIDE + koff];
    uint64_t ga = (uint64_t)(uintptr_t)(emb + n * DIM + koff);
    asm volatile("global_load_async_to_lds_b128 %0, %1, off"
                 :
                 : "v"(lds_off), "v"(ga)
                 : "memory");
  }

  // half squared norms of codes (from global; overlaps the async LDS fill)
  for (int n = tid; n < K_CODES; n += 256) {
    const float* e = emb + n * DIM;
    float s = 0.f;
#pragma unroll
    for (int d = 0; d < DIM; ++d) s += e[d] * e[d];
    e2s[n] = 0.5f * s;
  }

  asm volatile("s_wait_asynccnt 0x0" ::: "memory");
  __syncthreads();

  // ---- load this wave's 16 rows in WMMA f32 A layout (lane=M, vgpr pair=K) -
  // flat[n][d] = z_e[b*262144 + d*4096 + (n & 4095)]
  const float* zb = z_e + zbatch + hw_base + wave * 16 + l15;
  float a0[16], a1[16];
#pragma unroll
  for (int kk = 0; kk < 16; ++kk) {
    int k = 4 * kk + 2 * lh;
    a0[kk] = zb[(size_t)k * 4096];
    a1[kk] = zb[(size_t)(k + 1) * 4096];
  }

  // ---- score = x.e - 0.5||e||^2 via V_WMMA_F32_16X16X4_F32, argmax over K --
  float best[8];
  int bidx[8];
#pragma unroll
  for (int v = 0; v < 8; ++v) { best[v] = -3.4e38f; bidx[v] = 0; }

  for (int t = 0; t < K_CODES / 16; ++t) {
    const int n0 = t << 4;
    const float c0 = -e2s[n0 + l15];  // C layout: value depends only on column
    v8f C = {c0, c0, c0, c0, c0, c0, c0, c0};
#pragma unroll
    for (int kk = 0; kk < 16; ++kk) {
      int k = 4 * kk + 2 * lh;
      v2f A;
      A.x = a0[kk];
      A.y = a1[kk];
      v2f B = *(const v2f*)&eL[(n0 + l15) * LDS_STRIDE + k];  // ds_load_b64
      C = __builtin_amdgcn_wmma_f32_16x16x4_f32(false, A, false, B, (short)0,
                                                C, false, false);
    }
    const int code = n0 + l15;
#pragma unroll
    for (int v = 0; v < 8; ++v) {
      float s = C[v];
      if (s > best[v]) { best[v] = s; bidx[v] = code; }
    }
  }

  // ---- argmax across the 16 lanes of each half-wave (tie -> lowest index) --
#pragma unroll
  for (int v = 0; v < 8; ++v) {
    float bv = best[v];
    int bi = bidx[v];
#pragma unroll
    for (int m = 8; m >= 1; m >>= 1) {
      float ov = __shfl_xor(bv, m, 32);
      int oi = __shfl_xor(bi, m, 32);
      if (ov > bv || (ov == bv && oi < bi)) { bv = ov; bi = oi; }
    }
    if (l15 == 0) {
      int r = wave * 16 + v + 8 * lh;  // row within WG
      sIdx[r] = bi;
      atomicAdd(&counts[bi], 1u);
    }
  }
  __syncthreads();

  // ---- gather z_q, write NCHW output, accumulate SSE, write indices --------
  float sse = 0.f;
  for (int i = tid; i < ROWS_PER_WG * DIM; i += 256) {
    int r = i & 127;
    int d = i >> 7;
    int bi = sIdx[r];
    float q = eL[bi * LDS_STRIDE + d];
    size_t o = zbatch + (size_t)d * 4096 + hw_base + r;
    float ze = z_e[o];
    float df = q - ze;
    sse += df * df;
    out[o] = q;  // z_q_st == z_q numerically (straight-through)
    if (d == 0) out_idx[wg * ROWS_PER_WG + r] = (float)bi;
  }
#pragma unroll
  for (int m = 16; m >= 1; m >>= 1) sse += __shfl_xor(sse, m, 32);
  if (lane == 0) wsum[wave] = sse;
  __syncthreads();
  if (tid == 0) {
    float s = 0.f;
#pragma unroll
    for (int w2 = 0; w2 < 8; ++w2) s += wsum[w2];
    atomicAdd(sse_g, s);
  }
}

// ---------------------------------------------------------------------------
// finalize: loss = 0.25 * sse / numel ; perplexity = exp(-sum p log(p+1e-10))
// ---------------------------------------------------------------------------
__global__ void vq_finalize(const unsigned int* __restrict__ counts,
                            const float* __restrict__ sse_g,
                            float* __restrict__ out) {
  __shared__ float red[512];
  int t = threadIdx.x;
  float p = (float)counts[t] * (1.0f / (float)N_VECS);
  red[t] = -p * logf(p + 1e-10f);
  __syncthreads();
  for (int s = 256; s > 0; s >>= 1) {
    if (t < s) red[t] += red[t + s];
    __syncthreads();
  }
  if (t == 0) {
    out[4194304] = 0.25f * sse_g[0] * (1.0f / 4194304.0f);
    out[4194305] = expf(red[0]);
  }
}

// ---------------------------------------------------------------------------
extern "C" void kernel_launch(void* const* d_in, const int* in_sizes, int n_in,
                              void* d_out, int out_size, void* d_ws,
                              size_t ws_size, hipStream_t stream) {
  const float* z_e = (const float*)d_in[0];
  const float* emb = (const float*)d_in[1];
  float* out = (float*)d_out;
  float* ws = (float*)d_ws;
  float* sse_g = ws;
  unsigned int* counts = (unsigned int*)(ws + 1);

  vq_init<<<(1 + K_CODES + 255) / 256, 256, 0, stream>>>(ws);
  vq_main<<<N_VECS / ROWS_PER_WG, 256, 0, stream>>>(
      z_e, emb, out, out + 4194306, sse_g, counts);
  vq_finalize<<<1, 512, 0, stream>>>(counts, sse_g, out);
}